// CrossAttentionLayer_v2_14791867367637
// MI455X (gfx1250) — compile-verified
//
#include <hip/hip_runtime.h>
#include <hip/hip_bf16.h>
#include <stddef.h>

// Problem constants from the reference
#define B_  1024
#define T_  200
#define D_  512
#define FF_ 2048

typedef __attribute__((ext_vector_type(2))) float v2f;
typedef __attribute__((ext_vector_type(4))) float v4f;
typedef __attribute__((ext_vector_type(8))) float v8f;

// ---------------------------------------------------------------------------
// Kernel 1: S[b, :] = sum_t item_emb[b, t, :]
// Pure streaming reduction: 420 MB read, bandwidth bound (~18 us at 23.3 TB/s).
// Non-temporal loads (item_emb > 192 MB L2, read once) so weights/activations
// stay resident. 1 block per batch row, 128 threads x float4 covers D=512.
// ---------------------------------------------------------------------------
__global__ __launch_bounds__(128)
void sum_items_kernel(const float* __restrict__ item, float* __restrict__ S)
{
    const int b  = blockIdx.x;
    const int d4 = threadIdx.x;                       // 0..127 (float4 lane)
    const v4f* p = (const v4f*)(item + (size_t)b * T_ * D_) + d4;
    const int stride = D_ / 4;                        // 128 v4f per row

    v4f a0 = 0.0f, a1 = 0.0f;
    int t = 0;
    for (; t + 2 <= T_; t += 2) {
        v4f x0 = __builtin_nontemporal_load(p + (size_t)t * stride);
        v4f x1 = __builtin_nontemporal_load(p + (size_t)(t + 1) * stride);
        a0 += x0;
        a1 += x1;
    }
    for (; t < T_; ++t)
        a0 += __builtin_nontemporal_load(p + (size_t)t * stride);

    *((v4f*)(S + (size_t)b * D_) + d4) = a0 + a1;
}

// ---------------------------------------------------------------------------
// Kernel 2: generic f32 WMMA GEMM  C = epilogue(A[MxK] @ Bw[KxN])
//   epilogue: (+bias[col])? -> relu? -> (+resid[row,col])?
// 128 threads = 4 waves; block tile 64x64; wave w owns rows [16w,16w+16),
// all 64 cols as four 16x16 f32 accumulators (V_WMMA_F32_16X16X4_F32, wave32).
//
// Fragment layouts per CDNA5 ISA §7.12.2:
//   A (16x4):  VGPR j = A[row=lane&15][k = j + 2*(lane>>4)]
//   B (4x16):  VGPR j = B[k = j + 2*(lane>>4)][col = lane&15]
//   C/D:       VGPR j = C[row = j + 8*(lane>>4)][col = lane&15]
//
// LDS layouts chosen so every fragment is ONE aligned ds_load_b64:
//   As[64][18]      : stride 18 (even) -> (a0,a1) pair 8B-aligned;
//                     18r mod 64 distinct for r=0..15 -> conflict-free.
//   Bs[8][64][2]    : k-pair-major, Bs[p][col] = {B[2p][col], B[2p+1][col]}
//                     -> lane's v2f B fragment is contiguous, no mov glue.
// ---------------------------------------------------------------------------
__global__ __launch_bounds__(128)
void wmma_gemm_kernel(const float* __restrict__ A, const float* __restrict__ Bw,
                      const float* __restrict__ bias, const float* __restrict__ resid,
                      float* __restrict__ C, int M, int N, int K, int relu)
{
    __shared__ __align__(16) float As[64][18];
    __shared__ __align__(16) float Bs[8][64][2];

    const int tid      = threadIdx.x;
    const int wave     = tid >> 5;
    const int lane     = tid & 31;
    const int laneHalf = (lane >> 4) & 1;
    const int laneIdx  = lane & 15;

    const int m0 = blockIdx.y * 64;
    const int n0 = blockIdx.x * 64;

    v8f acc[4];
#pragma unroll
    for (int c = 0; c < 4; ++c)
#pragma unroll
        for (int j = 0; j < 8; ++j) acc[c][j] = 0.0f;

    for (int k0 = 0; k0 < K; k0 += 16) {
        // Stage A tile (64x16): 128 threads x 2 float4 loads, scalar LDS stores
        {
            const int r  = tid >> 2;            // 0..31
            const int c4 = (tid & 3) << 2;      // 0,4,8,12
#pragma unroll
            for (int p = 0; p < 2; ++p) {
                const int rr = r + p * 32;
                v4f v = *(const v4f*)(A + (size_t)(m0 + rr) * K + k0 + c4);
                As[rr][c4 + 0] = v[0];
                As[rr][c4 + 1] = v[1];
                As[rr][c4 + 2] = v[2];
                As[rr][c4 + 3] = v[3];
            }
        }
        // Stage B tile (16x64) into k-paired layout
        {
            const int r  = tid >> 4;            // 0..7
            const int c4 = (tid & 15) << 2;     // 0..60
#pragma unroll
            for (int p = 0; p < 2; ++p) {
                const int rr  = r + p * 8;      // 0..15
                const int pr  = rr >> 1;        // k-pair index
                const int par = rr & 1;         // even/odd slot
                v4f v = *(const v4f*)(Bw + (size_t)(k0 + rr) * N + n0 + c4);
                Bs[pr][c4 + 0][par] = v[0];
                Bs[pr][c4 + 1][par] = v[1];
                Bs[pr][c4 + 2][par] = v[2];
                Bs[pr][c4 + 3][par] = v[3];
            }
        }
        __syncthreads();

        const int arow = (wave << 4) + laneIdx;
        const float* arowp = &As[arow][2 * laneHalf];
#pragma unroll
        for (int kk = 0; kk < 16; kk += 4) {
            v2f a = *(const v2f*)(arowp + kk);                       // ds_load_b64
#pragma unroll
            for (int c = 0; c < 4; ++c) {
                v2f bfrag = *(const v2f*)(
                    &Bs[(kk >> 1) + laneHalf][(c << 4) + laneIdx][0]); // ds_load_b64
                acc[c] = __builtin_amdgcn_wmma_f32_16x16x4_f32(
                    false, a, false, bfrag, (short)0, acc[c], false, false);
            }
        }
        __syncthreads();
    }

    // Epilogue + store
    const int rbase = m0 + (wave << 4) + (laneHalf << 3);
#pragma unroll
    for (int c = 0; c < 4; ++c) {
        const int col = n0 + (c << 4) + laneIdx;
        const float bv = bias ? bias[col] : 0.0f;
#pragma unroll
        for (int j = 0; j < 8; ++j) {
            const int row = rbase + j;
            float v = acc[c][j] + bv;
            if (relu) v = fmaxf(v, 0.0f);
            if (resid) v += resid[(size_t)row * N + col];
            C[(size_t)row * N + col] = v;
        }
    }
}

// ---------------------------------------------------------------------------
// Launch. softmax(axis=1) over the singleton query axis == all-ones, so the
// attention output is just (sum_t item_emb) @ W_V; Q/K/W_Q/W_K/mask are dead.
// ---------------------------------------------------------------------------
extern "C" void kernel_launch(void* const* d_in, const int* in_sizes, int n_in,
                              void* d_out, int out_size, void* d_ws, size_t ws_size,
                              hipStream_t stream)
{
    const float* user_emb = (const float*)d_in[0];   // (B, D)
    const float* item_emb = (const float*)d_in[1];   // (B, T, D)
    // d_in[2] = W_Q, d_in[3] = W_K : dead code (softmax over singleton axis)
    const float* W_V   = (const float*)d_in[4];      // (D, D)
    const float* ff_W1 = (const float*)d_in[5];      // (D, FF)
    const float* ff_b1 = (const float*)d_in[6];      // (FF,)
    const float* ff_W2 = (const float*)d_in[7];      // (FF, D)
    const float* ff_b2 = (const float*)d_in[8];      // (D,)
    float* out = (float*)d_out;                      // (B, 1, D) flat

    // Workspace layout: 12 MB total
    float* S    = (float*)d_ws;                      // (B, D)  sum over T
    float* A1   = S  + (size_t)B_ * D_;              // (B, D)  attention out
    float* Hbuf = A1 + (size_t)B_ * D_;              // (B, FF) hidden

    // 1) S = sum_t item_emb  (the 420 MB streaming phase — the real cost)
    sum_items_kernel<<<B_, 128, 0, stream>>>(item_emb, S);

    // 2) A1 = S @ W_V                       (M=1024, N=512,  K=512)
    wmma_gemm_kernel<<<dim3(D_ / 64, B_ / 64), 128, 0, stream>>>(
        S, W_V, nullptr, nullptr, A1, B_, D_, D_, 0);

    // 3) H = relu(A1 @ ff_W1 + b1)          (M=1024, N=2048, K=512)
    wmma_gemm_kernel<<<dim3(FF_ / 64, B_ / 64), 128, 0, stream>>>(
        A1, ff_W1, ff_b1, nullptr, Hbuf, B_, FF_, D_, 1);

    // 4) out = H @ ff_W2 + b2 + user_emb    (M=1024, N=512,  K=2048)
    wmma_gemm_kernel<<<dim3(D_ / 64, B_ / 64), 128, 0, stream>>>(
        Hbuf, ff_W2, ff_b2, user_emb, out, B_, D_, FF_, 0);
}